// GATv2Attention_50929722196901
// MI455X (gfx1250) — compile-verified
//
#include <hip/hip_runtime.h>
#include <math.h>
#include <stdint.h>

#define B_ 16
#define H_ 8
#define N_ 256
#define D_ 32

#define ROWS_PER_BLOCK 32
#define ROWS_PER_WAVE  4
#define JCHUNKS        8   // 256 columns / 32 lanes

// ---- CDNA5 async global->LDS copies (ASYNCcnt-tracked DMA path) ----
__device__ __forceinline__ void async_load_b32(const void* gaddr, unsigned lds_off) {
  asm volatile("global_load_async_to_lds_b32 %0, %1, off"
               :: "v"(lds_off), "v"(gaddr) : "memory");
}
__device__ __forceinline__ void async_load_b128(const void* gaddr, unsigned lds_off) {
  asm volatile("global_load_async_to_lds_b128 %0, %1, off"
               :: "v"(lds_off), "v"(gaddr) : "memory");
}
__device__ __forceinline__ void wait_async0() {
  asm volatile("s_wait_asynccnt 0x0" ::: "memory");
}

__device__ __forceinline__ float fast_tanh(float x) {
#if __has_builtin(__builtin_amdgcn_tanhf)
  return __builtin_amdgcn_tanhf(x);          // gfx1250 v_tanh_f32 (TRANS)
#else
  // tanh(x) = 2/(1+exp(-2x)) - 1 ; exp(-2x) = exp2(-2*log2(e)*x)
  float e = __builtin_amdgcn_exp2f(-2.885390082f * x);
  return __builtin_amdgcn_rcpf(1.0f + e) * 2.0f - 1.0f;
#endif
}

__global__ __launch_bounds__(256) void gatv2_attn_kernel(
    const float* __restrict__ q, const float* __restrict__ k,
    const unsigned char* __restrict__ mask, const float* __restrict__ avec,
    float* __restrict__ out)
{
  __shared__ float kt[D_][N_];              // transposed K: kt[d][j] (32 KB)
  __shared__ float qs[ROWS_PER_BLOCK][D_];  // Q row slab (4 KB)
  __shared__ float as[D_];                  // attention vector for this head

  const int tid  = threadIdx.x;
  const int wave = tid >> 5;
  const int lane = tid & 31;

  const int bid  = blockIdx.x;   // B*H*8 = 1024 blocks
  const int slab = bid & 7;      // which 32-row slab
  const int bh   = bid >> 3;     // fused (b,h)
  const int b    = bh >> 3;      // / H_
  const int h    = bh & 7;       // % H_

  const float* kbase = k + (size_t)bh * N_ * D_;
  const float* qbase = q + ((size_t)bh * N_ + (size_t)slab * ROWS_PER_BLOCK) * D_;

  // ---------- async stage K (transposed scatter), Q slab, a ----------
  // K: lanes take consecutive j for a wave-uniform d -> conflict-free LDS writes.
#pragma unroll
  for (int m = 0; m < 4; ++m) {
    const int d = wave * 4 + m;
#pragma unroll
    for (int jc = 0; jc < JCHUNKS; ++jc) {
      const int j = lane + 32 * jc;
      async_load_b32(kbase + (size_t)j * D_ + d,
                     (unsigned)(uintptr_t)&kt[d][j]);
    }
  }
  // Q slab: 1024 floats, one b128 per thread, fully coalesced.
  async_load_b128(qbase + tid * 4, (unsigned)(uintptr_t)(&qs[0][0] + tid * 4));
  if (wave == 0) {
    async_load_b32(avec + h * D_ + lane, (unsigned)(uintptr_t)&as[lane]);
  }
  wait_async0();
  __syncthreads();

  // ---------- main compute: scores = sum_d a_d * silu(q+k) ----------
  const int r0 = wave * ROWS_PER_WAVE;   // first row (within slab) for this wave
  float acc[ROWS_PER_WAVE][JCHUNKS];
#pragma unroll
  for (int r = 0; r < ROWS_PER_WAVE; ++r)
#pragma unroll
    for (int c = 0; c < JCHUNKS; ++c) acc[r][c] = 0.0f;

  for (int d = 0; d < D_; ++d) {
    const float ad = as[d];              // LDS broadcast
    float qv[ROWS_PER_WAVE];
#pragma unroll
    for (int r = 0; r < ROWS_PER_WAVE; ++r)
      qv[r] = 0.5f * qs[r0 + r][d];      // fold the /2 for the tanh form
#pragma unroll
    for (int c = 0; c < JCHUNKS; ++c) {
      const float kv = 0.5f * kt[d][lane + 32 * c];  // conflict-free banks
#pragma unroll
      for (int r = 0; r < ROWS_PER_WAVE; ++r) {
        const float s = qv[r] + kv;          // x/2
        const float t = fast_tanh(s);        // 1 TRANS per element
        const float v = fmaf(s, t, s);       // silu(x) = s*(1 + tanh(s))
        acc[r][c] = fmaf(ad, v, acc[r][c]);
      }
    }
  }

  // ---------- mask + row softmax (wave32 shuffle reductions) ----------
  const size_t obase = ((size_t)bh * N_ + (size_t)slab * ROWS_PER_BLOCK) * N_;
  const unsigned char* mbase =
      mask + ((size_t)b * N_ + (size_t)slab * ROWS_PER_BLOCK) * (size_t)N_;
  const float NEG = -3.402823466e38f;

#pragma unroll
  for (int r = 0; r < ROWS_PER_WAVE; ++r) {
    const int row = r0 + r;
    const unsigned char* mrow = mbase + (size_t)row * N_;
    float s[JCHUNKS];
    float m = NEG;
#pragma unroll
    for (int c = 0; c < JCHUNKS; ++c) {
      const int j = lane + 32 * c;
      float v = acc[r][c];
      if (mrow[j]) v = NEG;
      s[c] = v;
      m = fmaxf(m, v);
    }
#pragma unroll
    for (int off = 16; off >= 1; off >>= 1)
      m = fmaxf(m, __shfl_xor(m, off, 32));

    float sum = 0.0f;
#pragma unroll
    for (int c = 0; c < JCHUNKS; ++c) {
      const float e = __expf(s[c] - m);      // v_exp_f32
      s[c] = e;
      sum += e;
    }
#pragma unroll
    for (int off = 16; off >= 1; off >>= 1)
      sum += __shfl_xor(sum, off, 32);

    const float inv = __builtin_amdgcn_rcpf(sum);
    float* orow = out + obase + (size_t)row * N_;
#pragma unroll
    for (int c = 0; c < JCHUNKS; ++c)
      orow[lane + 32 * c] = s[c] * inv;      // coalesced stores
  }
}

extern "C" void kernel_launch(void* const* d_in, const int* in_sizes, int n_in,
                              void* d_out, int out_size, void* d_ws, size_t ws_size,
                              hipStream_t stream) {
  (void)in_sizes; (void)n_in; (void)d_ws; (void)ws_size; (void)out_size;
  const float*         q    = (const float*)d_in[0];
  const float*         k    = (const float*)d_in[1];
  /* d_in[2] = scale, unused by the reference module */
  const unsigned char* mask = (const unsigned char*)d_in[3];
  const float*         avec = (const float*)d_in[4];
  float*               out  = (float*)d_out;

  dim3 grid(B_ * H_ * (N_ / ROWS_PER_BLOCK));   // 1024 blocks
  dim3 block(256);                               // 8 wave32s
  gatv2_attn_kernel<<<grid, block, 0, stream>>>(q, k, mask, avec, out);
}